// SLDS_51247549775962
// MI455X (gfx1250) — compile-verified
//
#include <hip/hip_runtime.h>
#include <math.h>

#define KC   8
#define HID  32
#define OBS  16
#define TLEN 1024
#define LOG2PI 1.8378770664093453f

typedef float v2f __attribute__((ext_vector_type(2)));
typedef float v8f __attribute__((ext_vector_type(8)));

#define V8Z {0.f,0.f,0.f,0.f,0.f,0.f,0.f,0.f}

__device__ __forceinline__ v8f wmma4(v2f a, v2f b, v8f c) {
  // D = A(16x4 f32) x B(4x16 f32) + C(16x16 f32)
  return __builtin_amdgcn_wmma_f32_16x16x4_f32(false, a, false, b, (short)0, c, false, false);
}

// ---- LDS layout (float offsets) ----
#define OFF_P    0        // 8*32*32   posterior cov P  (also T1 scratch)
#define OFF_B    8192     // 8*32*32   Pbar / Pm (prior cov)
#define OFF_PC   16384    // 8*32*16   PC = Pm @ C
#define OFF_S    20480    // 8*16*16   S, then L, then Linv
#define OFF_SI   22528    // 8*16*16   Sinv
#define OFF_KG   24576    // 8*32*16   Kalman gain
#define OFF_MP   28672    // 8*16*32   m_pred (rows 8..15 zero-padded)
#define OFF_DW   32768    // 8*8*32    sqrt(p)-weighted mean deviations
#define OFF_M    34816    // 16*32     posterior means (rows 8..15 zero)
#define OFF_MU   35328    // 8*32      prior / moment-matched means
#define OFF_LN   35584    // 8*8       normalized transition logits
#define OFF_PP   35648    // 8*8       responsibilities p
#define OFF_QP   35712    // 8*8       sqrt(p)
#define OFF_LW   35776    // 8         log weights
#define OFF_LD   35784    // 8         logdet(S)
#define OFF_INN  35792    // 8*16      innovation
#define OFF_SIV  35920    // 8*16      Sinv @ innov
#define OFF_Q    36048    // 32        exp(log_transition_noise)
#define OFF_R    36080    // 16        exp(log_obs_noise)
#define OFF_A    36096    // 8*32*32   LDS-staged transition matrices
#define OFF_C    44288    // 32*16     LDS-staged observation matrix
#define SMEM_FLOATS 44800 // 175 KB of 320 KB WGP LDS

// =====================  measurement update  =====================
__device__ __forceinline__ void obs_update(float* sm, const float* __restrict__ y, int tid) {
  const int w = tid >> 5, lane = tid & 31;
  const int half = lane >> 4, lr = lane & 15;
  float* sP  = sm + OFF_P;   float* sB  = sm + OFF_B;
  float* sPC = sm + OFF_PC;  float* sS  = sm + OFF_S;
  float* sSI = sm + OFF_SI;  float* sKG = sm + OFF_KG;
  float* sM  = sm + OFF_M;   float* sMU = sm + OFF_MU;
  float* sLW = sm + OFF_LW;  float* sLD = sm + OFF_LD;
  float* sIN = sm + OFF_INN; float* sSIv= sm + OFF_SIV;
  float* sR  = sm + OFF_R;   float* sC  = sm + OFF_C;

  // O1: PC[j] = Pm[j](sB) @ C   -- 16 jobs (j, mtile), K=32
  {
    const int j = w >> 1, a0 = (w & 1) * 16;
    v8f acc = V8Z;
#pragma unroll
    for (int ks = 0; ks < 8; ++ks) {
      const int k0 = ks * 4 + 2 * half;
      v2f a = *(const v2f*)&sB[(j * 32 + a0 + lr) * 32 + k0];
      v2f b; b.x = sC[k0 * 16 + lr]; b.y = sC[(k0 + 1) * 16 + lr];
      acc = wmma4(a, b, acc);
    }
#pragma unroll
    for (int r = 0; r < 8; ++r)
      sPC[(j * 32 + a0 + r + 8 * half) * 16 + lr] = acc[r];
  }
  __syncthreads();

  // O2: S[j] = C^T PC[j] + diag(R)  (waves 0-7)  |  innov (waves 8-11)
  if (w < 8) {
    const int j = w;
    v8f acc = V8Z;
#pragma unroll
    for (int ks = 0; ks < 8; ++ks) {
      const int k0 = ks * 4 + 2 * half;
      v2f a; a.x = sC[k0 * 16 + lr]; a.y = sC[(k0 + 1) * 16 + lr];
      v2f b; b.x = sPC[(j * 32 + k0) * 16 + lr]; b.y = sPC[(j * 32 + k0 + 1) * 16 + lr];
      acc = wmma4(a, b, acc);
    }
#pragma unroll
    for (int r = 0; r < 8; ++r) {
      const int m = r + 8 * half;
      sS[(j * 16 + m) * 16 + lr] = acc[r] + ((m == lr) ? sR[m] : 0.f);
    }
  } else if (w < 12) {
    const int jj = (w - 8) * 2 + half;
    float ym = 0.f;
#pragma unroll
    for (int h = 0; h < 32; ++h) ym += sMU[jj * 32 + h] * sC[h * 16 + lr];
    sIN[jj * 16 + lr] = y[lr] - ym;
  }
  __syncthreads();

  // O3: per-component Cholesky S = L L^T and Linv (waves 0-7, lane-parallel)
  if (w < 8) {
    const int j = w, r = lane & 15;
    float row[16];
#pragma unroll
    for (int c = 0; c < 16; ++c) row[c] = sS[(j * 16 + r) * 16 + c];
    float Lc[16];
    float logdet = 0.f;
#pragma unroll
    for (int c = 0; c < 16; ++c) {
      float piv = sqrtf(fmaxf(__shfl(row[c], c, 32), 1e-20f));
      logdet += 2.f * logf(piv);
      float l = (r == c) ? piv : ((r > c) ? row[c] / piv : 0.f);
      Lc[c] = l;
#pragma unroll
      for (int i = c + 1; i < 16; ++i) {
        float lic = __shfl(l, i, 32);
        row[i] -= l * lic;
      }
    }
#pragma unroll
    for (int c = 0; c < 16; ++c) sS[(j * 16 + r) * 16 + c] = Lc[c]; // L (upper = 0)
    if (lane == 0) sLD[j] = logdet;
    // Linv: lane r owns column r, forward substitution (masked, fully unrolled)
    float x[16];
#pragma unroll
    for (int i = 0; i < 16; ++i) x[i] = 0.f;
#pragma unroll
    for (int i = 0; i < 16; ++i) {
      float s = (i == r) ? 1.f : 0.f;
#pragma unroll
      for (int i2 = 0; i2 < 16; ++i2)
        if (i2 < i) s -= sS[(j * 16 + i) * 16 + i2] * x[i2];
      float xi = s / sS[(j * 16 + i) * 16 + i];
      x[i] = (i >= r) ? xi : 0.f;
    }
#pragma unroll
    for (int i = 0; i < 16; ++i) sS[(j * 16 + i) * 16 + r] = x[i]; // Linv
  }
  __syncthreads();

  // O4: Sinv = Linv^T Linv (waves 0-7), K=16
  if (w < 8) {
    const int j = w;
    v8f acc = V8Z;
#pragma unroll
    for (int ks = 0; ks < 4; ++ks) {
      const int k0 = ks * 4 + 2 * half;
      v2f a; a.x = sS[(j * 16 + k0) * 16 + lr]; a.y = sS[(j * 16 + k0 + 1) * 16 + lr];
      acc = wmma4(a, a, acc);
    }
#pragma unroll
    for (int r = 0; r < 8; ++r) sSI[(j * 16 + r + 8 * half) * 16 + lr] = acc[r];
  }
  __syncthreads();

  // O5: Kg = PC @ Sinv  -- 16 jobs, K=16
  {
    const int j = w >> 1, a0 = (w & 1) * 16;
    v8f acc = V8Z;
#pragma unroll
    for (int ks = 0; ks < 4; ++ks) {
      const int k0 = ks * 4 + 2 * half;
      v2f a = *(const v2f*)&sPC[(j * 32 + a0 + lr) * 16 + k0];
      v2f b; b.x = sSI[(j * 16 + k0) * 16 + lr]; b.y = sSI[(j * 16 + k0 + 1) * 16 + lr];
      acc = wmma4(a, b, acc);
    }
#pragma unroll
    for (int r = 0; r < 8; ++r)
      sKG[(j * 32 + a0 + r + 8 * half) * 16 + lr] = acc[r];
  }
  __syncthreads();

  // O6: Sinv@innov (waves 0-3) | m_new = mu + Kg innov (waves 4-11)
  if (w < 4) {
    const int jj = w * 2 + half;
    float s = 0.f;
#pragma unroll
    for (int o2 = 0; o2 < 16; ++o2) s += sSI[(jj * 16 + lr) * 16 + o2] * sIN[jj * 16 + o2];
    sSIv[jj * 16 + lr] = s;
  } else if (w < 12) {
    const int jj = w - 4;
    float v = sMU[jj * 32 + lane];
#pragma unroll
    for (int o = 0; o < 16; ++o) v += sKG[(jj * 32 + lane) * 16 + o] * sIN[jj * 16 + o];
    sM[jj * 32 + lane] = v;
  }
  __syncthreads();

  // O7: logw += ll (wave0) ; P_new = Pm - Kg PC^T (32 jobs, 2 per wave)
  if (w == 0 && lane < 8) {
    const int j = lane;
    float q = 0.f;
#pragma unroll
    for (int o = 0; o < 16; ++o) q += sIN[j * 16 + o] * sSIv[j * 16 + o];
    sLW[j] += -0.5f * ((float)OBS * LOG2PI + sLD[j] + q);
  }
#pragma unroll
  for (int it = 0; it < 2; ++it) {
    const int qd = w + it * 16;
    const int j = qd >> 2, a0 = ((qd >> 1) & 1) * 16, b0 = (qd & 1) * 16;
    v8f acc;
#pragma unroll
    for (int r = 0; r < 8; ++r) acc[r] = sB[(j * 32 + a0 + r + 8 * half) * 32 + b0 + lr];
#pragma unroll
    for (int ks = 0; ks < 4; ++ks) {
      const int k0 = ks * 4 + 2 * half;
      v2f a = *(const v2f*)&sKG[(j * 32 + a0 + lr) * 16 + k0];
      a = -a;
      v2f b = *(const v2f*)&sPC[(j * 32 + b0 + lr) * 16 + k0];
      acc = wmma4(a, b, acc);
    }
#pragma unroll
    for (int r = 0; r < 8; ++r)
      sP[(j * 32 + a0 + r + 8 * half) * 32 + b0 + lr] = acc[r];
  }
  __syncthreads();

  // O8: symmetrize P in place (pair-owner scheme, race-free)
  for (int idx = tid; idx < 8192; idx += 512) {
    const int j = idx >> 10, rem = idx & 1023, a = rem >> 5, b = rem & 31;
    if (b >= a) {
      float x = sP[(j * 32 + a) * 32 + b];
      float y2 = sP[(j * 32 + b) * 32 + a];
      float v = 0.5f * (x + y2);
      sP[(j * 32 + a) * 32 + b] = v;
      sP[(j * 32 + b) * 32 + a] = v;
    }
  }
  __syncthreads();
}

// =====================  predict + moment match  =====================
__device__ __forceinline__ void predict(float* sm, int tid) {
  const int w = tid >> 5, lane = tid & 31;
  const int half = lane >> 4, lr = lane & 15;
  float* sP  = sm + OFF_P;   float* sB  = sm + OFF_B;
  float* sMP = sm + OFF_MP;  float* sDW = sm + OFF_DW;
  float* sM  = sm + OFF_M;   float* sMU = sm + OFF_MU;
  float* sLN = sm + OFF_LN;  float* sPP = sm + OFF_PP;
  float* sQP = sm + OFF_QP;  float* sLW = sm + OFF_LW;
  float* sQ  = sm + OFF_Q;   float* sA  = sm + OFF_A;

  // D1: responsibilities (wave0) + m_pred[j] = m @ A[j] (16 jobs all waves, K=32)
  if (w == 0 && lane < 8) {
    const int jj = lane;
    float mx = -1e30f;
#pragma unroll
    for (int k = 0; k < 8; ++k) mx = fmaxf(mx, sLW[k] + sLN[k * 8 + jj]);
    float s = 0.f;
#pragma unroll
    for (int k = 0; k < 8; ++k) s += expf(sLW[k] + sLN[k * 8 + jj] - mx);
    float lw = mx + logf(s);
#pragma unroll
    for (int k = 0; k < 8; ++k) {
      float pp = expf(sLW[k] + sLN[k * 8 + jj] - lw);
      sPP[k * 8 + jj] = pp;
      sQP[k * 8 + jj] = sqrtf(pp);
    }
    sLW[jj] = lw;
  }
  {
    const int j = w >> 1, n0 = (w & 1) * 16;
    v8f acc = V8Z;
#pragma unroll
    for (int ks = 0; ks < 8; ++ks) {
      const int k0 = ks * 4 + 2 * half;
      v2f a = *(const v2f*)&sM[lr * 32 + k0];
      v2f b; b.x = sA[(j * 32 + k0) * 32 + n0 + lr]; b.y = sA[(j * 32 + k0 + 1) * 32 + n0 + lr];
      acc = wmma4(a, b, acc);
    }
#pragma unroll
    for (int r = 0; r < 8; ++r)
      sMP[(j * 16 + r + 8 * half) * 32 + n0 + lr] = acc[r];
  }
  __syncthreads();

  // D2: mu & weighted deviations (waves 0-7) ; Pbar = sum_k p P[k] (all threads, f2)
  if (w < 8) {
    const int j = w;
    float mu = 0.f;
#pragma unroll
    for (int k = 0; k < 8; ++k) mu += sPP[k * 8 + j] * sMP[(j * 16 + k) * 32 + lane];
    sMU[j * 32 + lane] = mu;
#pragma unroll
    for (int k = 0; k < 8; ++k)
      sDW[(j * 8 + k) * 32 + lane] = sQP[k * 8 + j] * (sMP[(j * 16 + k) * 32 + lane] - mu);
  }
#pragma unroll
  for (int it = 0; it < 8; ++it) {              // 4096 float2 outputs / 512 threads
    const int q = tid + it * 512;               // 0..4095
    const int j = q >> 9, hi = (q & 511) * 2;
    v2f s; s.x = 0.f; s.y = 0.f;
#pragma unroll
    for (int k = 0; k < 8; ++k) {
      const float pk = sPP[k * 8 + j];
      v2f pv = *(const v2f*)&sP[k * 1024 + hi];
      s.x += pk * pv.x; s.y += pk * pv.y;
    }
    *(v2f*)&sB[j * 1024 + hi] = s;
  }
  __syncthreads();

  // D3: T1[j] = A[j]^T Pbar[j]  -> sP (P is dead)  -- 32 jobs, K=32
#pragma unroll
  for (int it = 0; it < 2; ++it) {
    const int qd = w + it * 16;
    const int j = qd >> 2, a0 = ((qd >> 1) & 1) * 16, i0 = (qd & 1) * 16;
    v8f acc = V8Z;
#pragma unroll
    for (int ks = 0; ks < 8; ++ks) {
      const int k0 = ks * 4 + 2 * half;
      v2f a; a.x = sA[(j * 32 + k0) * 32 + a0 + lr]; a.y = sA[(j * 32 + k0 + 1) * 32 + a0 + lr];
      v2f b; b.x = sB[(j * 32 + k0) * 32 + i0 + lr]; b.y = sB[(j * 32 + k0 + 1) * 32 + i0 + lr];
      acc = wmma4(a, b, acc);
    }
#pragma unroll
    for (int r = 0; r < 8; ++r)
      sP[(j * 32 + a0 + r + 8 * half) * 32 + i0 + lr] = acc[r];
  }
  __syncthreads();

  // D4: Pm = T1 @ A + Dw^T Dw + diag(Q)  -> sB  -- 32 jobs, K=32 (+K=8 rank update)
#pragma unroll
  for (int it = 0; it < 2; ++it) {
    const int qd = w + it * 16;
    const int j = qd >> 2, a0 = ((qd >> 1) & 1) * 16, b0 = (qd & 1) * 16;
    v8f acc = V8Z;
#pragma unroll
    for (int ks = 0; ks < 8; ++ks) {
      const int k0 = ks * 4 + 2 * half;
      v2f a = *(const v2f*)&sP[(j * 32 + a0 + lr) * 32 + k0];
      v2f b; b.x = sA[(j * 32 + k0) * 32 + b0 + lr]; b.y = sA[(j * 32 + k0 + 1) * 32 + b0 + lr];
      acc = wmma4(a, b, acc);
    }
#pragma unroll
    for (int ks = 0; ks < 2; ++ks) {
      const int k0 = ks * 4 + 2 * half;
      v2f a; a.x = sDW[(j * 8 + k0) * 32 + a0 + lr]; a.y = sDW[(j * 8 + k0 + 1) * 32 + a0 + lr];
      v2f b; b.x = sDW[(j * 8 + k0) * 32 + b0 + lr]; b.y = sDW[(j * 8 + k0 + 1) * 32 + b0 + lr];
      acc = wmma4(a, b, acc);
    }
#pragma unroll
    for (int r = 0; r < 8; ++r) {
      const int m = a0 + r + 8 * half, n = b0 + lr;
      sB[(j * 32 + m) * 32 + n] = acc[r] + ((m == n) ? sQ[m] : 0.f);
    }
  }
  __syncthreads();
}

// =====================  persistent kernel  =====================
extern "C" __global__ void __launch_bounds__(512, 1)
slds_filter_kernel(const float* __restrict__ data,
                   const float* __restrict__ tlog,
                   const float* __restrict__ Ag,
                   const float* __restrict__ logQ,
                   const float* __restrict__ Cg,
                   const float* __restrict__ logR,
                   float* __restrict__ out) {
  extern __shared__ float sm[];
  const int tid = threadIdx.x;
  float* sB  = sm + OFF_B;  float* sM  = sm + OFF_M;
  float* sMU = sm + OFF_MU; float* sLN = sm + OFF_LN;
  float* sLW = sm + OFF_LW; float* sQ  = sm + OFF_Q;
  float* sR  = sm + OFF_R;  float* sA  = sm + OFF_A;
  float* sC  = sm + OFF_C;

  // ---- init: stage A & C, prior cov = I, means = 0, constants ----
  for (int q = tid; q < 8192; q += 512) {
    sA[q] = Ag[q];
    const int hi = q & 1023;
    sB[q] = ((hi >> 5) == (hi & 31)) ? 1.f : 0.f;
  }
  sC[tid] = Cg[tid];                   // 32*16 == 512 == blockDim
  sM[tid] = 0.f;                       // 16*32 == 512 == blockDim
  if (tid < 256) sMU[tid] = 0.f;
  if (tid < 32) sQ[tid] = expf(logQ[tid]);
  if (tid < 16) sR[tid] = expf(logR[tid]);
  if (tid < 8) {
    float mx = -1e30f;
#pragma unroll
    for (int j = 0; j < 8; ++j) mx = fmaxf(mx, tlog[tid * 8 + j]);
    float s = 0.f;
#pragma unroll
    for (int j = 0; j < 8; ++j) s += expf(tlog[tid * 8 + j] - mx);
    const float lse = mx + logf(s);
#pragma unroll
    for (int j = 0; j < 8; ++j) sLN[tid * 8 + j] = tlog[tid * 8 + j] - lse;
  }
  __syncthreads();
  if (tid < 8) sLW[tid] = sLN[tid];    // logw0 = L[0,:]
  __syncthreads();

  obs_update(sm, data, tid);
  for (int t = 1; t < TLEN; ++t) {
    if ((tid >> 5) == 0) __builtin_prefetch(data + t * OBS, 0, 1); // next y row
    predict(sm, tid);
    obs_update(sm, data + t * OBS, tid);
  }

  if (tid == 0) {
    float mx = -1e30f;
#pragma unroll
    for (int k = 0; k < 8; ++k) mx = fmaxf(mx, sLW[k]);
    float s = 0.f;
#pragma unroll
    for (int k = 0; k < 8; ++k) s += expf(sLW[k] - mx);
    out[0] = mx + logf(s);
  }
}

extern "C" void kernel_launch(void* const* d_in, const int* in_sizes, int n_in,
                              void* d_out, int out_size, void* d_ws, size_t ws_size,
                              hipStream_t stream) {
  (void)in_sizes; (void)n_in; (void)out_size; (void)d_ws; (void)ws_size;
  const float* data = (const float*)d_in[0];   // (1024,16)
  const float* tlog = (const float*)d_in[1];   // (8,8)
  const float* A    = (const float*)d_in[2];   // (8,32,32)
  const float* logQ = (const float*)d_in[3];   // (32,)
  const float* Cg   = (const float*)d_in[4];   // (32,16)
  const float* logR = (const float*)d_in[5];   // (16,)
  float* out = (float*)d_out;
  const size_t smem = (size_t)SMEM_FLOATS * sizeof(float); // ~175 KB of 320 KB WGP LDS
  slds_filter_kernel<<<1, 512, smem, stream>>>(data, tlog, A, logQ, Cg, logR, out);
}